// FourierMMLayer_59734405152992
// MI455X (gfx1250) — compile-verified
//
#include <hip/hip_runtime.h>
#include <hip/hip_bf16.h>

typedef float v2f __attribute__((ext_vector_type(2)));
typedef float v8f __attribute__((ext_vector_type(8)));
typedef int   i32x4 __attribute__((ext_vector_type(4)));

#define BATCH   32
#define SEQ     512
#define HID     768
#define DH_PLANE (768 * 768)          // one f32 plane of Dh
#define DS_CAT   (512 * 1024)         // [Ds_r | -Ds_i]
#define T_BATCH  (1024 * 768)         // stacked [T_r; T_i] per batch

#if __has_builtin(__builtin_amdgcn_global_load_async_to_lds_b128)
#define HAS_ASYNC_LDS 1
#else
#define HAS_ASYNC_LDS 0
#endif

__device__ __forceinline__ void wait_asynccnt0() {
#if __has_builtin(__builtin_amdgcn_s_wait_asynccnt)
    __builtin_amdgcn_s_wait_asynccnt(0);
#else
    asm volatile("s_wait_asynccnt 0x0" ::: "memory");
#endif
}

#if HAS_ASYNC_LDS
__device__ __forceinline__ void async_copy_b128(const float* gsrc, float* ldst) {
    __builtin_amdgcn_global_load_async_to_lds_b128(
        (__attribute__((address_space(1))) i32x4*)gsrc,
        (__attribute__((address_space(3))) i32x4*)ldst,
        /*offset=*/0, /*cpol=*/0);
}
#endif

// ---------------------------------------------------------------------------
// complex128 -> f32 plane conversion
// ---------------------------------------------------------------------------
__global__ void convert_dh(const double* __restrict__ dh, float* __restrict__ dhp) {
    int idx = blockIdx.x * blockDim.x + threadIdx.x;   // over 768*768
    if (idx >= DH_PLANE) return;
    double re = dh[2 * idx + 0];
    double im = dh[2 * idx + 1];
    dhp[idx]            = (float)re;   // Dh_r plane
    dhp[DH_PLANE + idx] = (float)im;   // Dh_i plane
}

__global__ void convert_ds(const double* __restrict__ ds, float* __restrict__ dsc) {
    int idx = blockIdx.x * blockDim.x + threadIdx.x;   // over 512*512
    if (idx >= SEQ * SEQ) return;
    int n = idx / SEQ;
    int i = idx % SEQ;
    double re = ds[2 * idx + 0];
    double im = ds[2 * idx + 1];
    dsc[n * 1024 + i]        =  (float)re;   // Ds_r
    dsc[n * 1024 + 512 + i]  = -(float)im;   // -Ds_i (so stage2 is one GEMM)
}

// ---------------------------------------------------------------------------
// 128x128-tile fp32 WMMA GEMM core, double-buffered LDS pipeline.
//   256 threads = 8 waves (wave32), waves tiled 2(M) x 4(N), each wave a
//   64x32 patch = 4x2 tiles of v_wmma_f32_16x16x4_f32.
// ---------------------------------------------------------------------------
#define BK      32
#define SA_LD   36        // 128 x 36 f32 (padded: conflict-free A-frag reads)
#define SB_LD   132       // 32 x 132 f32

__device__ __forceinline__ void wmma_compute(const float* sAb, const float* sBb,
                                             v8f acc[4][2],
                                             int waveM, int waveN, int l15, int khalf)
{
    #pragma unroll
    for (int kk = 0; kk < BK; kk += 4) {
        const int ko = kk + khalf;          // per-half-wave K offset (+0 / +2)
        v2f afrag[4];
        #pragma unroll
        for (int mi = 0; mi < 4; ++mi) {
            int r = waveM * 64 + mi * 16 + l15;
            afrag[mi] = *(const v2f*)(&sAb[r * SA_LD + ko]);   // even: 8B aligned
        }
        v2f bfrag[2];
        #pragma unroll
        for (int ni = 0; ni < 2; ++ni) {
            int c = waveN * 32 + ni * 16 + l15;
            v2f bf;
            bf.x = sBb[(ko + 0) * SB_LD + c];
            bf.y = sBb[(ko + 1) * SB_LD + c];
            bfrag[ni] = bf;
        }
        #pragma unroll
        for (int mi = 0; mi < 4; ++mi)
            #pragma unroll
            for (int ni = 0; ni < 2; ++ni)
                acc[mi][ni] = __builtin_amdgcn_wmma_f32_16x16x4_f32(
                    false, afrag[mi], false, bfrag[ni],
                    (short)0, acc[mi][ni], false, false);
    }
}

__device__ __forceinline__ void gemm_tile_f32(
    const float* __restrict__ A, int lda,
    const float* __restrict__ B, int ldb,
    float* __restrict__ C, int ldc,
    int K, int bm, int bn)
{
    __shared__ float sA[2][128 * SA_LD];
    __shared__ float sB[2][BK * SB_LD];

    const int tid   = threadIdx.x;
    const int lane  = tid & 31;
    const int wave  = tid >> 5;
    const int waveM = wave >> 2;          // 0..1  -> M offset *64
    const int waveN = wave & 3;           // 0..3  -> N offset *32
    const int l15   = lane & 15;
    const int khalf = (lane >> 4) << 1;

    // per-thread float4 slots: A tile 128x32 (8 f4/row), B tile 32x128 (32 f4/row)
    int aRow[4], aCol[4], bRow[4], bCol[4];
    #pragma unroll
    for (int i = 0; i < 4; ++i) {
        int id = tid + i * 256;
        aRow[i] = id >> 3;  aCol[i] = (id & 7) * 4;
        bRow[i] = id >> 5;  bCol[i] = (id & 31) * 4;
    }

    v8f acc[4][2] = {};
    const int nk = K / BK;

#if HAS_ASYNC_LDS
    // ---------------- async global->LDS pipeline (ASYNCcnt) ----------------
    #pragma unroll
    for (int i = 0; i < 4; ++i) {
        async_copy_b128(A + (size_t)(bm + aRow[i]) * lda + aCol[i], &sA[0][aRow[i] * SA_LD + aCol[i]]);
        async_copy_b128(B + (size_t)bRow[i] * ldb + bn + bCol[i],   &sB[0][bRow[i] * SB_LD + bCol[i]]);
    }
    wait_asynccnt0();
    __syncthreads();

    for (int it = 0; it < nk; ++it) {
        const int cur = it & 1;
        if (it + 1 < nk) {
            const int nxt = cur ^ 1;
            const int k1  = (it + 1) * BK;
            #pragma unroll
            for (int i = 0; i < 4; ++i) {
                async_copy_b128(A + (size_t)(bm + aRow[i]) * lda + k1 + aCol[i],
                                &sA[nxt][aRow[i] * SA_LD + aCol[i]]);
                async_copy_b128(B + (size_t)(k1 + bRow[i]) * ldb + bn + bCol[i],
                                &sB[nxt][bRow[i] * SB_LD + bCol[i]]);
            }
        }
        wmma_compute(sA[cur], sB[cur], acc, waveM, waveN, l15, khalf);
        if (it + 1 < nk) wait_asynccnt0();
        __syncthreads();
    }
#else
    // ------------- register-staged pipeline (fallback path) ----------------
    float4 rA[4], rB[4];
    #pragma unroll
    for (int i = 0; i < 4; ++i) {
        rA[i] = *(const float4*)(A + (size_t)(bm + aRow[i]) * lda + aCol[i]);
        rB[i] = *(const float4*)(B + (size_t)bRow[i] * ldb + bn + bCol[i]);
    }
    #pragma unroll
    for (int i = 0; i < 4; ++i) {
        *(float4*)(&sA[0][aRow[i] * SA_LD + aCol[i]]) = rA[i];
        *(float4*)(&sB[0][bRow[i] * SB_LD + bCol[i]]) = rB[i];
    }
    __syncthreads();

    for (int it = 0; it < nk; ++it) {
        const int cur = it & 1;
        if (it + 1 < nk) {
            const int k1 = (it + 1) * BK;
            #pragma unroll
            for (int i = 0; i < 4; ++i) {
                rA[i] = *(const float4*)(A + (size_t)(bm + aRow[i]) * lda + k1 + aCol[i]);
                rB[i] = *(const float4*)(B + (size_t)(k1 + bRow[i]) * ldb + bn + bCol[i]);
            }
        }
        wmma_compute(sA[cur], sB[cur], acc, waveM, waveN, l15, khalf);
        if (it + 1 < nk) {
            const int nxt = cur ^ 1;
            #pragma unroll
            for (int i = 0; i < 4; ++i) {
                *(float4*)(&sA[nxt][aRow[i] * SA_LD + aCol[i]]) = rA[i];
                *(float4*)(&sB[nxt][bRow[i] * SB_LD + bCol[i]]) = rB[i];
            }
        }
        __syncthreads();
    }
#endif

    // ---- write back: C/D layout — VGPR v: rows v (lanes 0-15) / v+8 (16-31) ----
    const int rhalf = (lane >> 4) << 3;
    #pragma unroll
    for (int mi = 0; mi < 4; ++mi) {
        #pragma unroll
        for (int ni = 0; ni < 2; ++ni) {
            int colg = bn + waveN * 32 + ni * 16 + l15;
            #pragma unroll
            for (int v = 0; v < 8; ++v) {
                int rowg = bm + waveM * 64 + mi * 16 + v + rhalf;
                C[(size_t)rowg * ldc + colg] = acc[mi][ni][v];
            }
        }
    }
}

// ---------------------------------------------------------------------------
// Stage 1: T[b][ri*512 + i][j] = (X_b @ DhPlane[ri])[i][j]
// ---------------------------------------------------------------------------
__global__ __launch_bounds__(256, 2)
void gemm_stage1(const float* __restrict__ X,
                 const float* __restrict__ DhP,
                 float* __restrict__ T)
{
    const int z  = blockIdx.z;
    const int b  = z >> 1;
    const int ri = z & 1;
    const float* A  = X   + (size_t)b  * SEQ * HID;          // 512x768
    const float* Bm = DhP + (size_t)ri * DH_PLANE;           // 768x768
    float*       Cp = T   + (size_t)b * T_BATCH + (size_t)ri * SEQ * HID;
    gemm_tile_f32(A, HID, Bm, HID, Cp, HID, HID,
                  blockIdx.y * 128, blockIdx.x * 128);
}

// ---------------------------------------------------------------------------
// Stage 2: out_b = DsCat (512x1024) @ T_b (1024x768)
// ---------------------------------------------------------------------------
__global__ __launch_bounds__(256, 2)
void gemm_stage2(const float* __restrict__ DsCat,
                 const float* __restrict__ T,
                 float* __restrict__ Out)
{
    const int b = blockIdx.z;
    gemm_tile_f32(DsCat, 1024,
                  T   + (size_t)b * T_BATCH, HID,
                  Out + (size_t)b * SEQ * HID, HID,
                  1024, blockIdx.y * 128, blockIdx.x * 128);
}

// ---------------------------------------------------------------------------
extern "C" void kernel_launch(void* const* d_in, const int* in_sizes, int n_in,
                              void* d_out, int out_size, void* d_ws, size_t ws_size,
                              hipStream_t stream) {
    (void)in_sizes; (void)n_in; (void)out_size; (void)ws_size;

    const float*  x  = (const float*)d_in[0];          // [32,512,768] f32
    const double* ds = (const double*)d_in[1];         // [512,512] complex128
    const double* dh = (const double*)d_in[2];         // [768,768] complex128
    float* out = (float*)d_out;

    float* ws  = (float*)d_ws;
    float* dhp = ws;                                   // 2 * 768*768
    float* dsc = dhp + 2 * DH_PLANE;                   // 512*1024
    float* tc  = dsc + DS_CAT;                         // 32 * 1024*768

    convert_dh<<<(DH_PLANE + 255) / 256, 256, 0, stream>>>(dh, dhp);
    convert_ds<<<(SEQ * SEQ + 255) / 256, 256, 0, stream>>>(ds, dsc);

    gemm_stage1<<<dim3(HID / 128, SEQ / 128, BATCH * 2), 256, 0, stream>>>(x, dhp, tc);
    gemm_stage2<<<dim3(HID / 128, SEQ / 128, BATCH),     256, 0, stream>>>(dsc, tc, out);
}